// GenTargets_17403207483863
// MI455X (gfx1250) — compile-verified
//
#include <hip/hip_runtime.h>

// FCOS target assignment for gfx1250 (MI455X) — single fused kernel.
// All 5 pyramid levels decoded per-thread via power-of-two arithmetic.
// Per-block: async-DMA (ASYNCcnt) this batch's 100 boxes into LDS, then
// precompute per-box invariants (cx, cy, area-or-INF with class validity
// folded in, class-as-float) into an 8-float LDS record. Main loop reads two
// ds_load_b128 broadcasts per box and does ~20 VALU ops per test.

#ifndef __has_builtin
#define __has_builtin(x) 0
#endif

#if __has_builtin(__builtin_amdgcn_global_load_async_to_lds_b128)
#define ASYNC_MODE 2
#elif __has_builtin(__builtin_amdgcn_global_load_async_to_lds_b32)
#define ASYNC_MODE 1
#else
#define ASYNC_MODE 0
#endif

#define NBOX     100
#define BLOCK    256
#define INF_AREA 1e10f
#define P_TOTAL  17064   // 12800 + 3200 + 800 + 208 + 56

// Builtin signatures (per hipcc diagnostics):
//   b32 : (AS1 int* src, AS3 int* dst, imm, imm)
//   b128: (AS1 int __vector(4)* src, AS3 int __vector(4)* dst, imm, imm)
typedef int v4i __attribute__((__vector_size__(16)));
typedef __attribute__((address_space(1))) int* gintp;
typedef __attribute__((address_space(3))) int* lintp;
typedef __attribute__((address_space(1))) v4i* gv4p;
typedef __attribute__((address_space(3))) v4i* lv4p;

__device__ __forceinline__ void wait_asynccnt0() {
#if __has_builtin(__builtin_amdgcn_s_wait_asynccnt)
    __builtin_amdgcn_s_wait_asynccnt(0);
#else
    asm volatile("s_wait_asynccnt 0" ::: "memory");
#endif
}

__global__ __launch_bounds__(BLOCK) void fcos_targets_kernel(
    const float* __restrict__ boxes,    // (B, 100, 4)
    const int*   __restrict__ classes,  // (B, 100)
    float* __restrict__ cls_out,        // (B, P_TOTAL)
    float* __restrict__ cnt_out,        // (B, P_TOTAL)
    float* __restrict__ reg_out)        // (B, P_TOTAL, 4)
{
    __shared__ __align__(16) float sraw[NBOX * 4];   // raw boxes (async-DMA target)
    __shared__ __align__(16) float sder[NBOX * 8];   // bx0,by0,bx1,by1,cx,cy,areaInf,clsF

    const int b   = blockIdx.y;
    const int tid = threadIdx.x;

    const float* gbox = boxes   + (size_t)b * (NBOX * 4);
    const int*   gcls = classes + (size_t)b * NBOX;

    // ---- stage raw boxes into LDS ----
#if ASYNC_MODE == 2
    if (tid < NBOX) {
        __builtin_amdgcn_global_load_async_to_lds_b128(
            (gv4p)(gbox + 4 * tid), (lv4p)(sraw + 4 * tid), 0, 0);
    }
    wait_asynccnt0();          // own box record now visible to this thread
#elif ASYNC_MODE == 1
    for (int i = tid; i < NBOX * 4; i += BLOCK) {
        __builtin_amdgcn_global_load_async_to_lds_b32(
            (gintp)(gbox + i), (lintp)(sraw + i), 0, 0);
    }
    wait_asynccnt0();
    __syncthreads();           // strided staging: need all waves' data
#else
    for (int i = tid; i < NBOX * 4; i += BLOCK) sraw[i] = gbox[i];
    __syncthreads();
#endif

    // ---- precompute per-box invariants (thread m owns box m) ----
    if (tid < NBOX) {
        const float bx0 = sraw[4 * tid + 0];
        const float by0 = sraw[4 * tid + 1];
        const float bx1 = sraw[4 * tid + 2];
        const float by1 = sraw[4 * tid + 3];
        const int   c   = gcls[tid];
        const float area = (bx1 - bx0) * (by1 - by0);
        float* rec = &sder[8 * tid];
        rec[0] = bx0;
        rec[1] = by0;
        rec[2] = bx1;
        rec[3] = by1;
        rec[4] = (bx0 + bx1) * 0.5f;                 // cx
        rec[5] = (by0 + by1) * 0.5f;                 // cy
        rec[6] = (c >= 0) ? area : INF_AREA;         // validity folded into area
        rec[7] = (float)c;                           // class as float
    }
    __syncthreads();

    const int gid = blockIdx.x * BLOCK + tid;        // flat location over all levels
    if (gid >= P_TOTAL) return;

    // ---- per-thread level decode (all params are powers of two) ----
    const int lvl = (gid >= 12800) + (gid >= 16000) + (gid >= 16800) + (gid >= 17008);
    const int off = (lvl == 0) ? 0
                  : (lvl == 1) ? 12800
                  : (lvl == 2) ? 16000
                  : (lvl == 3) ? 16800
                               : 17008;
    const int   log2W  = 7 - lvl;                    // W = 128 >> lvl
    const int   pix    = gid - off;
    const int   yi     = pix >> log2W;
    const int   xi     = pix & ((1 << log2W) - 1);
    const float stride = (float)(8 << lvl);
    const float minr   = (lvl == 0) ? -1.0f : (float)(32 << lvl);
    const float maxr   = (lvl == 4) ? 1000000000.0f : (float)(64 << lvl);

    const float sx   = (float)xi * stride + stride * 0.5f;
    const float sy   = (float)yi * stride + stride * 0.5f;
    const float crad = stride * 1.5f;

    float best_area = INF_AREA;
    int   best_m    = 0;

#pragma unroll 4
    for (int m = 0; m < NBOX; ++m) {
        const float4 A = *(const float4*)&sder[8 * m];      // bx0,by0,bx1,by1
        const float4 D = *(const float4*)&sder[8 * m + 4];  // cx,cy,areaInf,clsF

        const float l = sx - A.x;
        const float t = sy - A.y;
        const float r = A.z - sx;
        const float d = A.w - sy;

        const float offmin = fminf(fminf(l, t), fminf(r, d));
        const float offmax = fmaxf(fmaxf(l, t), fmaxf(r, d));

        // max(|dx|,|dy|) < crad  <=>  |dx| < crad && |dy| < crad  (abs = free src mod)
        const bool ok = (offmin >= 0.0f) & (offmax >= minr) & (offmax <= maxr)
                      & (fabsf(sx - D.x) < crad) & (fabsf(sy - D.y) < crad);

        const float area = ok ? D.z : INF_AREA;             // D.z already INF if class<0
        if (area < best_area) { best_area = area; best_m = m; }  // first-min wins
    }

    const bool pos = best_area < INF_AREA;

    // recompute ltrb + class for the selected box
    const float4 A = *(const float4*)&sder[8 * best_m];
    const float l = sx - A.x, t = sy - A.y, r = A.z - sx, d = A.w - sy;

    float  cls_v = 0.0f;
    float  cnt_v = -1.0f;
    float4 reg_v = make_float4(-1.0f, -1.0f, -1.0f, -1.0f);
    if (pos) {
        cls_v = sder[8 * best_m + 7];
        const float lr_min = fminf(l, r);
        const float lr_max = fmaxf(fmaxf(l, r), 1e-5f);
        const float tb_min = fminf(t, d);
        const float tb_max = fmaxf(fmaxf(t, d), 1e-5f);
        const float ratio  = (lr_min * tb_min) / (lr_max * tb_max + 1e-10f);
        cnt_v = sqrtf(fmaxf(ratio, 1e-12f));
        reg_v = make_float4(l, t, r, d);
    }

    const size_t loc = (size_t)b * (size_t)P_TOTAL + (size_t)gid;
    cls_out[loc] = cls_v;
    cnt_out[loc] = cnt_v;
    *(float4*)(reg_out + loc * 4) = reg_v;  // 16B-aligned, coalesced
}

extern "C" void kernel_launch(void* const* d_in, const int* in_sizes, int n_in,
                              void* d_out, int out_size, void* d_ws, size_t ws_size,
                              hipStream_t stream) {
    // Inputs (setup_inputs order): feat0..feat4 (shape-only, unused),
    // gt_boxes = d_in[5] (B,100,4) f32, classes = d_in[6] (B,100) i32.
    const float* boxes   = (const float*)d_in[5];
    const int*   classes = (const int*)  d_in[6];
    const int    B       = in_sizes[6] / NBOX;   // classes flat = B*100

    float* cls_out = (float*)d_out;                      // (B, P)
    float* cnt_out = cls_out + (size_t)B * P_TOTAL;      // (B, P)
    float* reg_out = cnt_out + (size_t)B * P_TOTAL;      // (B, P, 4)

    const int gx = (P_TOTAL + BLOCK - 1) / BLOCK;        // 67
    fcos_targets_kernel<<<dim3(gx, B), dim3(BLOCK), 0, stream>>>(
        boxes, classes, cls_out, cnt_out, reg_out);
}